// SAGELayer_6528350290008
// MI455X (gfx1250) — compile-verified
//
#include <hip/hip_runtime.h>

// GraphSAGE layer for MI455X (gfx1250, wave32, WMMA).
//  d_in: 0 nfeats[N,64] f32, 1 efeats[E,64] f32, 2 src[E] i32, 3 dst[E] i32,
//        4 W_msg[128,64] f32, 5 b_msg[64] f32, 6 W_apply[128,64] f32, 7 b_apply[64] f32
//  d_out: h[N,64] f32   (also used as the msum accumulator between kernels 2 and 3)
//  d_ws : deg[N] f32

#define WAVES_PER_BLOCK 8
#define TILE_M 16
#define KDIM 128
#define NDIM 64
#define WSTRIDE 132   // padded row stride (floats): bank = (4*row + k) mod 64 -> conflict-free b64 LDS reads

typedef float v2f __attribute__((ext_vector_type(2)));
typedef float v8f __attribute__((ext_vector_type(8)));

// One wave computes a 16x64 = A(16x128) @ W(128x64) + bias tile with
// 32 k-steps x 4 n-tiles of v_wmma_f32_16x16x4_f32.
// sA: 16 rows, stride WSTRIDE.  sW: n-major transposed weights, 64 rows, stride WSTRIDE.
__device__ __forceinline__ void wmma_gemm_16x128_by_128x64(
    const float* sA, const float* sW, const float* __restrict__ bias,
    int lane, v8f acc[4])
{
    const int nloc = lane & 15;
    const int hi   = lane >> 4;

#pragma unroll
    for (int t = 0; t < 4; ++t) {
        float bv = bias[t * 16 + nloc];   // C/D: lane -> column, so bias splats across the 8 row-regs
        v8f c;
#pragma unroll
        for (int i = 0; i < 8; ++i) c[i] = bv;
        acc[t] = c;
    }

    // A f32 16x4 layout: lanes 0-15 hold (row=lane, K=k,k+1), lanes 16-31 hold (row=lane-16, K=k+2,k+3).
    const float* aBase = sA + (lane & 15) * WSTRIDE + hi * 2;
    // B f32 4x16 layout mirrors A: lower lanes = lower K pair, v0/v1 advance K within the pair.
    const float* wBase = sW + nloc * WSTRIDE + hi * 2;

#pragma unroll 4
    for (int k = 0; k < KDIM; k += 4) {
        v2f a;
        a.x = aBase[k];
        a.y = aBase[k + 1];
#pragma unroll
        for (int t = 0; t < 4; ++t) {
            const float* wp = wBase + t * 16 * WSTRIDE + k;
            v2f b;
            b.x = wp[0];
            b.y = wp[1];
            acc[t] = __builtin_amdgcn_wmma_f32_16x16x4_f32(
                false, a, false, b, (short)0, acc[t], false, false);
        }
    }
}

__global__ __launch_bounds__(256) void sage_zero_kernel(float* __restrict__ out, float* __restrict__ deg,
                                                        long nOut, long nDeg)
{
    long i = (long)blockIdx.x * blockDim.x + threadIdx.x;
    long stride = (long)gridDim.x * blockDim.x;
    for (long j = i; j < nOut; j += stride) out[j] = 0.0f;
    for (long j = i; j < nDeg; j += stride) deg[j] = 0.0f;
}

__global__ __launch_bounds__(256) void sage_edge_kernel(
    const float* __restrict__ nfeats, const float* __restrict__ efeats,
    const int* __restrict__ src, const int* __restrict__ dst,
    const float* __restrict__ W_msg, const float* __restrict__ b_msg,
    float* __restrict__ msum, float* __restrict__ deg, int E)
{
    __shared__ float sW[NDIM * WSTRIDE];                        // ~33.8 KB, W_msg transposed [n][k]
    __shared__ float sA[WAVES_PER_BLOCK][TILE_M * WSTRIDE];     // ~67.6 KB, per-wave A tiles
    __shared__ int   sDst[WAVES_PER_BLOCK][TILE_M];

    const int tid  = threadIdx.x;
    const int lane = tid & 31;
    const int w    = tid >> 5;

    // Stage transposed weights once per block (L2-resident source).
    for (int idx = tid; idx < KDIM * NDIM; idx += blockDim.x) {
        int k = idx >> 6;      // 0..127
        int n = idx & 63;      // 0..63
        sW[n * WSTRIDE + k] = W_msg[idx];
    }

    const int  tile  = blockIdx.x * WAVES_PER_BLOCK + w;
    const long e0    = (long)tile * TILE_M;
    const bool valid = e0 < E;                 // wave-uniform
    const bool full  = (e0 + TILE_M) <= E;     // wave-uniform (always true for E % 16 == 0)

    if (valid) {
        if (lane < TILE_M) {
            long e = e0 + lane;
            sDst[w][lane] = (e < E) ? dst[e] : -1;
        }
        // Two lanes per edge row: half 0 gathers nfeats[src], half 1 copies efeats.
        int  r    = lane >> 1;
        int  half = lane & 1;
        long e    = e0 + r;
        float* arow = &sA[w][r * WSTRIDE + half * 64];
        if (e < E) {
            const float4* p = (half == 0)
                ? (const float4*)(nfeats + (long)src[e] * 64)
                : (const float4*)(efeats + e * 64);
#pragma unroll
            for (int j = 0; j < 16; ++j) ((float4*)arow)[j] = p[j];
        } else {
            float4 z = make_float4(0.f, 0.f, 0.f, 0.f);
#pragma unroll
            for (int j = 0; j < 16; ++j) ((float4*)arow)[j] = z;
        }
    }
    __syncthreads();   // block-uniform

    if (valid) {
        v8f acc[4];
        wmma_gemm_16x128_by_128x64(sA[w], sW, b_msg, lane, acc);

        const int nloc = lane & 15;
        const int hi   = lane >> 4;

        if (full) {
            // Fast path: no per-element guards. Preload the 8 row indices once,
            // one 64-bit base address per row, 4 column atomics via imm offsets.
            int drow[8];
#pragma unroll
            for (int v = 0; v < 8; ++v) drow[v] = sDst[w][v + 8 * hi];
#pragma unroll
            for (int v = 0; v < 8; ++v) {
                float* p = msum + (long)drow[v] * 64 + nloc;
#pragma unroll
                for (int t = 0; t < 4; ++t)
                    atomicAdd(p + t * 16, acc[t][v]);
            }
            if (lane < TILE_M) atomicAdd(&deg[sDst[w][lane]], 1.0f);
        } else {
#pragma unroll
            for (int v = 0; v < 8; ++v) {
                int row = v + 8 * hi;              // C/D layout: VGPR v -> row v + 8*hi
                int d = sDst[w][row];
                if (d >= 0) {
                    float* p = msum + (long)d * 64 + nloc;
#pragma unroll
                    for (int t = 0; t < 4; ++t)
                        atomicAdd(p + t * 16, acc[t][v]);
                }
            }
            if (lane < TILE_M) {
                int d = sDst[w][lane];
                if (d >= 0) atomicAdd(&deg[d], 1.0f);
            }
        }
    }
}

__global__ __launch_bounds__(256) void sage_apply_kernel(
    const float* __restrict__ nfeats,
    const float* __restrict__ W_apply, const float* __restrict__ b_apply,
    float* __restrict__ out /* in: msum, out: h (rows disjoint per tile) */,
    const float* __restrict__ deg, int N)
{
    __shared__ float sW[NDIM * WSTRIDE];
    __shared__ float sA[WAVES_PER_BLOCK][TILE_M * WSTRIDE];

    const int tid  = threadIdx.x;
    const int lane = tid & 31;
    const int w    = tid >> 5;

    for (int idx = tid; idx < KDIM * NDIM; idx += blockDim.x) {
        int k = idx >> 6;
        int n = idx & 63;
        sW[n * WSTRIDE + k] = W_apply[idx];
    }

    const int  tile  = blockIdx.x * WAVES_PER_BLOCK + w;
    const long n0    = (long)tile * TILE_M;
    const bool valid = n0 < N;
    const bool full  = (n0 + TILE_M) <= N;

    if (valid) {
        int  r    = lane >> 1;
        int  half = lane & 1;
        long node = n0 + r;
        float* arow = &sA[w][r * WSTRIDE + half * 64];
        if (node < N) {
            if (half == 0) {
                const float4* p = (const float4*)(nfeats + node * 64);
#pragma unroll
                for (int j = 0; j < 16; ++j) ((float4*)arow)[j] = p[j];
            } else {
                float dv = deg[node];
                float scale = (dv > 0.0f) ? (1.0f / dv) : 0.0f;   // mean; 0 for isolated nodes
                const float* mp = out + node * 64;                 // read msum before this tile is overwritten
#pragma unroll
                for (int j = 0; j < 64; ++j) arow[j] = mp[j] * scale;
            }
        } else {
#pragma unroll
            for (int j = 0; j < 64; ++j) arow[j] = 0.0f;
        }
    }
    __syncthreads();

    if (valid) {
        v8f acc[4];
        wmma_gemm_16x128_by_128x64(sA[w], sW, b_apply, lane, acc);

        const int nloc = lane & 15;
        const int hi   = lane >> 4;

        if (full) {
#pragma unroll
            for (int v = 0; v < 8; ++v) {
                long node = n0 + (v + 8 * hi);
                float* p = out + node * 64 + nloc;
#pragma unroll
                for (int t = 0; t < 4; ++t)
                    p[t * 16] = fmaxf(acc[t][v], 0.0f);
            }
        } else {
#pragma unroll
            for (int v = 0; v < 8; ++v) {
                long node = n0 + (v + 8 * hi);
                if (node < N) {
                    float* p = out + node * 64 + nloc;
#pragma unroll
                    for (int t = 0; t < 4; ++t)
                        p[t * 16] = fmaxf(acc[t][v], 0.0f);
                }
            }
        }
    }
}

extern "C" void kernel_launch(void* const* d_in, const int* in_sizes, int n_in,
                              void* d_out, int out_size, void* d_ws, size_t ws_size,
                              hipStream_t stream)
{
    const float* nfeats  = (const float*)d_in[0];
    const float* efeats  = (const float*)d_in[1];
    const int*   src     = (const int*)d_in[2];
    const int*   dst     = (const int*)d_in[3];
    const float* W_msg   = (const float*)d_in[4];
    const float* b_msg   = (const float*)d_in[5];
    const float* W_apply = (const float*)d_in[6];
    const float* b_apply = (const float*)d_in[7];
    float* out = (float*)d_out;
    float* deg = (float*)d_ws;   // N floats

    const int N = in_sizes[0] / NDIM;
    const int E = in_sizes[2];

    // 1) zero accumulators (d_out doubles as msum; deg in workspace)
    sage_zero_kernel<<<2048, 256, 0, stream>>>(out, deg, (long)N * NDIM, (long)N);

    // 2) edge messages + scatter-add
    int etiles  = (E + TILE_M - 1) / TILE_M;
    int eblocks = (etiles + WAVES_PER_BLOCK - 1) / WAVES_PER_BLOCK;
    sage_edge_kernel<<<eblocks, 256, 0, stream>>>(nfeats, efeats, src, dst,
                                                  W_msg, b_msg, out, deg, E);

    // 3) mean + apply + relu (in place on d_out)
    int ntiles  = (N + TILE_M - 1) / TILE_M;
    int nblocks = (ntiles + WAVES_PER_BLOCK - 1) / WAVES_PER_BLOCK;
    sage_apply_kernel<<<nblocks, 256, 0, stream>>>(nfeats, W_apply, b_apply,
                                                   out, deg, N);
}